// Agent_56899726737926
// MI455X (gfx1250) — compile-verified
//
#include <hip/hip_runtime.h>
#include <stdint.h>

// ---------------------------------------------------------------------------
// Workload: graph-batched categorical entropy / log-prob (memory-bound).
// No matmul structure -> WMMA inapplicable; the CDNA5 feature that applies is
// the Tensor Data Mover (strided tile -> LDS) used in the per-graph softmax.
// ---------------------------------------------------------------------------

typedef unsigned int u32;
typedef __attribute__((ext_vector_type(4))) unsigned int u32x4;
typedef __attribute__((ext_vector_type(8))) int          i32x8;
typedef __attribute__((ext_vector_type(4))) int          i32x4;

#define ENTRIES_PER_THREAD 8
#define MAX_LOCS_PER_GRAPH 4096
#define MAX_BF 1152   // B*(F+1) LDS accumulator budget for kgmean

// ---------------- workspace zeroing ----------------------------------------
__global__ void Agent_kzero(float* __restrict__ p, int n) {
    int i = blockIdx.x * blockDim.x + threadIdx.x;
    int stride = gridDim.x * blockDim.x;
    for (; i < n; i += stride) p[i] = 0.0f;
}

// ---------------- per-entry gather + sorted segment-sum ---------------------
// vals[i] = sum_f (type==1 ? logits : edge_vf)[id[i], f];
// loc_scores[loc[i]] += vals[i]   (entry_loc sorted -> run-length combine)
__global__ void Agent_kentries(const float* __restrict__ logits,
                               const float* __restrict__ edge_vf,
                               const int* __restrict__ etype,
                               const int* __restrict__ eid,
                               const int* __restrict__ eloc,
                               float* __restrict__ loc_scores,
                               int NE, int F) {
    long long base = (long long)(blockIdx.x * blockDim.x + threadIdx.x) * ENTRIES_PER_THREAD;
    if (base >= NE) return;
    long long end = base + ENTRIES_PER_THREAD;
    if (end > NE) end = NE;

    int   cur = eloc[base];
    float acc = 0.0f;
    for (long long i = base; i < end; ++i) {
        int id = eid[i];
        const float* row = (etype[i] == 1) ? (logits + (size_t)id * F)
                                           : (edge_vf + (size_t)id * F);
        float v;
        if (F == 8) {
            const float4* r4 = (const float4*)row;   // rows are 32B-aligned
            float4 a = r4[0], b = r4[1];
            v = (a.x + a.y) + (a.z + a.w) + (b.x + b.y) + (b.z + b.w);
        } else {
            v = 0.0f;
            for (int f = 0; f < F; ++f) v += row[f];
        }
        int l = eloc[i];
        if (l != cur) {
            atomicAdd(loc_scores + cur, acc);
            cur = l;
            acc = v;
        } else {
            acc += v;
        }
    }
    atomicAdd(loc_scores + cur, acc);
}

// ---------------- scatter-mean of logits over graphs ------------------------
__global__ void Agent_kgmean(const float* __restrict__ logits,
                             const int* __restrict__ node_batch,
                             float* __restrict__ g_sum,     // [B*F]
                             float* __restrict__ counts,    // [B]
                             int N, int B, int F) {
    __shared__ float acc[MAX_BF];               // [B*F] sums then [B] counts
    const int nslots = B * F + B;
    for (int i = threadIdx.x; i < nslots; i += blockDim.x) acc[i] = 0.0f;
    __syncthreads();

    int stride = gridDim.x * blockDim.x;
    for (int i = blockIdx.x * blockDim.x + threadIdx.x; i < N; i += stride) {
        int b = node_batch[i];
        const float* row = logits + (size_t)i * F;
        if (F == 8) {
            const float4* r4 = (const float4*)row;
            float4 a = r4[0], c = r4[1];
            float* dst = acc + b * 8;
            atomicAdd(dst + 0, a.x); atomicAdd(dst + 1, a.y);
            atomicAdd(dst + 2, a.z); atomicAdd(dst + 3, a.w);
            atomicAdd(dst + 4, c.x); atomicAdd(dst + 5, c.y);
            atomicAdd(dst + 6, c.z); atomicAdd(dst + 7, c.w);
        } else {
            for (int f = 0; f < F; ++f) atomicAdd(acc + b * F + f, row[f]);
        }
        atomicAdd(acc + B * F + b, 1.0f);
    }
    __syncthreads();
    for (int i = threadIdx.x; i < B * F; i += blockDim.x) atomicAdd(g_sum + i, acc[i]);
    for (int i = threadIdx.x; i < B;     i += blockDim.x) atomicAdd(counts + i, acc[B * F + i]);
}

// ---------------- per-graph softmax stats (TDM strided tile -> LDS) ---------
// Graph g owns locations g, g+B, g+2B, ...  (loc_graph = arange(L) % B).
__global__ void Agent_ksoftmax(const float* __restrict__ loc_scores,
                               const float* __restrict__ g_sum,
                               const float* __restrict__ counts,
                               float* __restrict__ lse,       // [B]
                               float* __restrict__ out,       // out[B+g] = entropy
                               int locsPerG, int B, int F) {
    const int g   = blockIdx.x;
    const int tid = threadIdx.x;
    __shared__ float smem[MAX_LOCS_PER_GRAPH];
    __shared__ float red[256];
    __shared__ float sS[2];   // sS[0] = m (g_mean slot), sS[1] = M

    if (tid == 0) {
        float c = fmaxf(counts[g], 1.0f);
        float s = 0.0f;
        for (int f = 0; f < F; ++f) s += g_sum[g * F + f];
        sS[0] = (s / c) / (float)F;
    }

    const bool useTdm = (locsPerG <= MAX_LOCS_PER_GRAPH);
    if (useTdm && tid < warpSize) {   // one wave issues the TDM descriptor
        u32 lds_off = (u32)(size_t)&smem[0];                    // LDS byte offset
        unsigned long long ga =
            (unsigned long long)(size_t)(loc_scores + g);       // tile start
        u32x4 g0;
        g0[0] = 1u;                                   // count=1, user descriptor
        g0[1] = lds_off;                              // lds_addr
        g0[2] = (u32)(ga & 0xFFFFFFFFu);              // global_addr[31:0]
        g0[3] = (u32)((ga >> 32) & 0x01FFFFFFu)       // global_addr[56:32]
              | (2u << 30);                           // type = 2 ("image")
        i32x8 g1;
        g1[0] = (int)(2u << 16);                      // wg_mask=0, data_size=4B
        g1[1] = (int)(((u32)B & 0xFFFFu) << 16);      // tensor_dim0 (low 16)
        g1[2] = (int)(((u32)locsPerG & 0xFFFFu) << 16); // tensor_dim1 (low 16)
        g1[3] = (int)(1u << 16);                      // tile_dim0 = 1
        g1[4] = (int)(u32)locsPerG;                   // tile_dim1
        g1[5] = (int)(u32)B;                          // tensor_dim0_stride = B
        g1[6] = 0;                                    // dim1_stride (unused, 2D)
        g1[7] = 0;
        i32x4 zz4; zz4[0] = 0; zz4[1] = 0; zz4[2] = 0; zz4[3] = 0;
        i32x8 zz8; zz8[0] = 0; zz8[1] = 0; zz8[2] = 0; zz8[3] = 0;
                   zz8[4] = 0; zz8[5] = 0; zz8[6] = 0; zz8[7] = 0;
        __builtin_amdgcn_tensor_load_to_lds(g0, g1, zz4, zz4, zz8, 0);
        __builtin_amdgcn_s_wait_tensorcnt(0);
    }
    __syncthreads();

    // pass 1: max over this graph's scores
    float mx = -INFINITY;
    for (int k = tid; k < locsPerG; k += blockDim.x) {
        float s = useTdm ? smem[k] : loc_scores[(size_t)g + (size_t)k * B];
        mx = fmaxf(mx, s);
    }
    red[tid] = mx; __syncthreads();
    for (int s = 128; s > 0; s >>= 1) {
        if (tid < s) red[tid] = fmaxf(red[tid], red[tid + s]);
        __syncthreads();
    }
    if (tid == 0) sS[1] = fmaxf(red[0], sS[0]);
    __syncthreads();
    const float M = sS[1];

    // pass 2: Z = sum e^{s-M}, P = sum s*e^{s-M}
    float zsum = 0.0f, psum = 0.0f;
    for (int k = tid; k < locsPerG; k += blockDim.x) {
        float s = useTdm ? smem[k] : loc_scores[(size_t)g + (size_t)k * B];
        float e = expf(s - M);
        zsum += e;
        psum += s * e;
    }
    red[tid] = zsum; __syncthreads();
    for (int s = 128; s > 0; s >>= 1) {
        if (tid < s) red[tid] += red[tid + s];
        __syncthreads();
    }
    float Zr = red[0]; __syncthreads();
    red[tid] = psum; __syncthreads();
    for (int s = 128; s > 0; s >>= 1) {
        if (tid < s) red[tid] += red[tid + s];
        __syncthreads();
    }
    if (tid == 0) {
        float m  = sS[0];
        float em = expf(m - M);
        float Z  = Zr + em;
        float P  = red[0] + m * em;
        float l  = logf(Z) + M;
        lse[g]       = l;
        out[B + g]   = l - P / Z;    // entropy row of the [2,B] output
    }
}

// ---------------- log_probs for chosen locations ----------------------------
__global__ void Agent_kaction(const float* __restrict__ loc_scores,
                              const float* __restrict__ lse,
                              const int* __restrict__ loc_graph,
                              const int* __restrict__ action_loc,
                              float* __restrict__ out, int B) {
    int b = blockIdx.x * blockDim.x + threadIdx.x;
    if (b < B) {
        int al = action_loc[b];
        out[b] = loc_scores[al] - lse[loc_graph[al]];
    }
}

// ---------------------------------------------------------------------------
extern "C" void kernel_launch(void* const* d_in, const int* in_sizes, int n_in,
                              void* d_out, int out_size, void* d_ws, size_t ws_size,
                              hipStream_t stream) {
    const float* logits     = (const float*)d_in[0];
    const float* edge_vf    = (const float*)d_in[1];
    const int*   node_batch = (const int*)d_in[2];
    const int*   entry_type = (const int*)d_in[3];
    const int*   entry_id   = (const int*)d_in[4];
    const int*   entry_loc  = (const int*)d_in[5];
    const int*   loc_graph  = (const int*)d_in[6];
    const int*   action_loc = (const int*)d_in[7];

    const int N  = in_sizes[2];
    const int NE = in_sizes[3];
    const int L  = in_sizes[6];
    const int B  = in_sizes[7];
    const int F  = in_sizes[0] / N;
    const int locsPerG = L / B;

    float* ws         = (float*)d_ws;
    float* loc_scores = ws;                 // [L]
    float* g_sum      = loc_scores + L;     // [B*F]
    float* counts     = g_sum + (size_t)B * F; // [B]
    float* lse        = counts + B;         // [B]
    float* out        = (float*)d_out;      // [2*B]: log_probs then entropy

    // 1) zero accumulators
    const int zn = L + B * F + B + B;
    Agent_kzero<<<(zn + 255) / 256, 256, 0, stream>>>(ws, zn);

    // 2) entry gathers -> sorted segment-sum into loc_scores
    {
        long long threads = (NE + ENTRIES_PER_THREAD - 1) / ENTRIES_PER_THREAD;
        int blocks = (int)((threads + 255) / 256);
        Agent_kentries<<<blocks, 256, 0, stream>>>(logits, edge_vf, entry_type,
                                                   entry_id, entry_loc,
                                                   loc_scores, NE, F);
    }

    // 3) scatter-mean of logits over graphs (LDS-staged atomics)
    Agent_kgmean<<<512, 256, 0, stream>>>(logits, node_batch, g_sum, counts,
                                          N, B, F);

    // 4) per-graph softmax stats: TDM strided tile into LDS, then reduce
    Agent_ksoftmax<<<B, 256, 0, stream>>>(loc_scores, g_sum, counts, lse, out,
                                          locsPerG, B, F);

    // 5) log_probs
    Agent_kaction<<<1, 256, 0, stream>>>(loc_scores, lse, loc_graph, action_loc,
                                         out, B);
}